// lenet_mnist_mvm_26817775796394
// MI455X (gfx1250) — compile-verified
//
#include <hip/hip_runtime.h>

// ---------------------------------------------------------------------------
// Quantized LeNet forward, fully fused per layer:
//   conv5x5(pad2) -> +bias -> fxp(24,12) -> relu -> maxpool2x2 -> fxp(16,12)
// as one implicit-GEMM kernel on V_WMMA_F32_16X16X4_F32 (exact fp32 math,
// matching the fp32 JAX reference with Q12 fixed-point quantization).
//
// Each wave owns a 4x4 spatial block x 16 output channels. With the WMMA
// C-layout (VGPR i: lanes 0-15 -> M=i, lanes 16-31 -> M=8+i; N = lane&15),
// every 2x2 pool window lives inside one lane's accumulator registers, so
// relu+quant+pool is fused in-register and pooled output is stored directly.
// Next-layer input quantization fxp(16,12) is applied in the same epilogue
// (exact: fxp16 is monotonic, so fxp16(max(..)) == reference order), which
// makes downstream inner loops pure load+WMMA.
//
// Weight B-panels are staged into LDS with the gfx1250 async data mover
// (GLOBAL_LOAD_ASYNC_TO_LDS_B128, ASYNCcnt) overlapped with the im2col
// gather, so the K-loop reads both operands from LDS (ds_load_b64).
// ---------------------------------------------------------------------------

typedef __attribute__((ext_vector_type(2))) float v2f;
typedef __attribute__((ext_vector_type(8))) float v8f;

// fxp(x, bits=16, frac=12): round-to-nearest-even on 2^-12 grid, clamp 16-bit
__device__ __forceinline__ float fxp16q(float x) {
    float r = rintf(x * 4096.0f);
    r = fminf(fmaxf(r, -32768.0f), 32767.0f);
    return r * 0.000244140625f;   // 2^-12
}
// fxp(x, bits=24, frac=12): accumulator quantization
__device__ __forceinline__ float fxp24q(float x) {
    float r = rintf(x * 4096.0f);
    r = fminf(fmaxf(r, -8388608.0f), 8388607.0f);
    return r * 0.000244140625f;
}

// ---------------------------------------------------------------------------
// Weight prep: quantize (16,12) and lay out as pair-interleaved B panels:
// element (k, n) of n-tile nt lives at  nt*KPAD*16 + ((k>>1)*16 + n)*2 + (k&1)
// so one lane's B fragment (rows kk, kk+1, column n) is one float2 load.
// Weight source is [COUT][K] row-major (OIHW flattened / wfc as-is).
// ---------------------------------------------------------------------------
template <int K, int COUT, int KPAD>
__global__ void prep_w_kernel(const float* __restrict__ w, float* __restrict__ Bq)
{
    constexpr int NT = (COUT + 15) / 16;
    int idx = blockIdx.x * blockDim.x + threadIdx.x;
    int total = NT * KPAD * 16;
    if (idx >= total) return;
    int nt  = idx / (KPAD * 16);
    int rem = idx - nt * (KPAD * 16);
    int k   = rem >> 4;
    int n   = rem & 15;
    int ng  = nt * 16 + n;
    float v = 0.0f;
    if (k < K && ng < COUT) v = fxp16q(w[ng * K + k]);
    Bq[(size_t)nt * KPAD * 16 + (((size_t)(k >> 1) * 16 + n) << 1) + (k & 1)] = v;
}

// ---------------------------------------------------------------------------
// Fused conv5x5(pad2) + bias + fxp24 + relu + maxpool2 + fxp16 kernel.
//   grid.x = (H/4)*(W/4) spatial blocks, grid.y = batch, grid.z = n-tiles
//   block  = 32 threads (one wave)
// QUANT_A: quantize input fxp(16,12) on load (layer 1 only; later layers'
// inputs are pre-quantized by the previous epilogue).
// ---------------------------------------------------------------------------
template <int CIN, int COUT, int H, int W, int KPAD, bool QUANT_A>
__global__ __launch_bounds__(32)
void conv_pool_wmma_kernel(const float* __restrict__ in,
                           const float* __restrict__ Bq,
                           const float* __restrict__ bias,
                           float* __restrict__ out)
{
    constexpr int K  = CIN * 25;
    constexpr int HW = H * W;
    constexpr int BW = W / 4;              // blocks per row
    const int lane = threadIdx.x;          // 0..31
    const int tile = blockIdx.x;           // 4x4 spatial block index
    const int img  = blockIdx.y;           // batch index
    const int nt   = blockIdx.z;           // n-tile

    __shared__ __align__(16) float As[16 * KPAD];  // im2col A tile
    __shared__ __align__(16) float Bs[KPAD * 16];  // weight B panel

    const float* inb = in + (size_t)img * CIN * HW;
    const float* Bn  = Bq + (size_t)nt * KPAD * 16;

    // Kick off the async B-panel copy into LDS (ASYNCcnt-tracked; overlaps
    // with the im2col gather below). 16 bytes per lane per step.
    {
        const unsigned long long gB = (unsigned long long)Bn;
        const unsigned int ldsB = (unsigned int)(unsigned long long)(void*)Bs;
        for (int ofs = lane * 16; ofs < KPAD * 16 * 4; ofs += 32 * 16) {
            unsigned long long ga = gB + (unsigned)ofs;
            unsigned int la = ldsB + (unsigned)ofs;
            asm volatile("global_load_async_to_lds_b128 %0, %1, off"
                         :: "v"(la), "v"(ga)
                         : "memory");
        }
    }

    // Pull this image's activations toward the WGP (global_prefetch_b8).
    __builtin_prefetch((const char*)inb + lane * 128, 0, 1);

    const int by4 = (tile / BW) * 4;
    const int bx4 = (tile - (tile / BW) * BW) * 4;

    // Cooperative im2col gather. 16*KPAD is a multiple of 32 for all layers,
    // so every lane runs the same trip count (EXEC all-ones at the WMMAs).
    for (int idx = lane; idx < 16 * KPAD; idx += 32) {
        int m = idx / KPAD;
        int k = idx - m * KPAD;
        float v = 0.0f;
        if (k < K) {
            int oy = by4 + (m >> 2);
            int ox = bx4 + (m & 3);
            int c  = k / 25, rs = k - c * 25;
            int r  = rs / 5, s  = rs - r * 5;
            int iy = oy - 2 + r, ix = ox - 2 + s;
            if (iy >= 0 && iy < H && ix >= 0 && ix < W) {
                float t = inb[(c * H + iy) * W + ix];
                v = QUANT_A ? fxp16q(t) : t;
            }
        }
        As[idx] = v;
    }

    // Wait for the async weight copy, then synchronize LDS views.
    asm volatile("s_wait_asynccnt 0x0" ::: "memory");
    __syncthreads();

    // Fragment indexing for V_WMMA_F32_16X16X4_F32 (wave32):
    //   A 16x4: lanes 0-15 hold K={0,1}, lanes 16-31 hold K={2,3}, M = lane&15
    //   B 4x16: same K split, N = lane&15
    const int m  = lane & 15;
    const int n  = lane & 15;
    const int kh = (lane >> 4) << 1;

    v8f acc = {};
    for (int k0 = 0; k0 < KPAD; k0 += 4) {
        int kk = k0 + kh;                                   // even
        float2 ap = *(const float2*)&As[m * KPAD + kk];     // ds_load_b64
        float2 bp = *(const float2*)&Bs[((kk >> 1) * 16 + n) << 1];
        v2f a; a.x = ap.x; a.y = ap.y;
        v2f b; b.x = bp.x; b.y = bp.y;
        acc = __builtin_amdgcn_wmma_f32_16x16x4_f32(
                  false, a, false, b, (short)0, acc, false, false);
    }

    // Fused epilogue: bias + fxp24 + relu, then in-register 2x2 max pool,
    // then next-layer input quantization fxp16. Each lane's 8 accumulator
    // values cover two complete pool windows.
    const int ng = nt * 16 + n;
    if (ng < COUT) {
        const float bv = bias[ng];
        float v[8];
#pragma unroll
        for (int i = 0; i < 8; ++i)
            v[i] = fmaxf(fxp24q(acc[i] + bv), 0.0f);        // quant + relu
        float q0 = fxp16q(fmaxf(fmaxf(v[0], v[1]), fmaxf(v[4], v[5])));
        float q1 = fxp16q(fmaxf(fmaxf(v[2], v[3]), fmaxf(v[6], v[7])));
        constexpr int Ho = H / 2, Wo = W / 2;
        const int prow = (by4 >> 1) + ((lane < 16) ? 0 : 1);
        float2 st; st.x = q0; st.y = q1;
        *(float2*)&out[(((size_t)img * COUT + ng) * Ho + prow) * Wo + (bx4 >> 1)] = st;
    }
}

// ---------------------------------------------------------------------------
// FC layer: out[b,0:10] = fxp24( h[b,:] @ Wfc^T + bfc )
// h is already fxp(16,12)-quantized by the layer-3 epilogue, so the inner
// loop is pure b64 loads + WMMA (codegen-verified: fully pipelined).
// One wave per 16-image tile.
// ---------------------------------------------------------------------------
__global__ __launch_bounds__(32)
void fc_wmma_kernel(const float* __restrict__ h,
                    const float* __restrict__ Bq,
                    const float* __restrict__ bias,
                    float* __restrict__ out)
{
    constexpr int K = 512;
    const int lane = threadIdx.x;
    const int tile = blockIdx.x;           // 16-image tile

    const int m  = lane & 15;
    const int n  = lane & 15;
    const int kh = (lane >> 4) << 1;
    const float* hm = h + (size_t)(tile * 16 + m) * K;

    __builtin_prefetch((const char*)hm, 0, 1);

    v8f acc = {};
    for (int k0 = 0; k0 < K; k0 += 4) {
        int kk = k0 + kh;
        float2 ap = *(const float2*)&hm[kk];
        float2 bp = *(const float2*)&Bq[(((size_t)(kk >> 1) * 16 + n) << 1)];
        v2f a; a.x = ap.x; a.y = ap.y;
        v2f b; b.x = bp.x; b.y = bp.y;
        acc = __builtin_amdgcn_wmma_f32_16x16x4_f32(
                  false, a, false, b, (short)0, acc, false, false);
    }

    if (n < 10) {
        const float bv = bias[n];
        const int mbase = (lane < 16) ? 0 : 8;
#pragma unroll
        for (int i = 0; i < 8; ++i) {
            int img = tile * 16 + mbase + i;
            out[(size_t)img * 10 + n] = fxp24q(acc[i] + bv);
        }
    }
}

// ---------------------------------------------------------------------------
// Host-side launch
// ---------------------------------------------------------------------------
extern "C" void kernel_launch(void* const* d_in, const int* in_sizes, int n_in,
                              void* d_out, int out_size, void* d_ws, size_t ws_size,
                              hipStream_t stream)
{
    const float* x   = (const float*)d_in[0];   // [B,1,32,32]
    const float* w1  = (const float*)d_in[1];   // [6,1,5,5]
    const float* b1  = (const float*)d_in[2];
    const float* w2  = (const float*)d_in[3];   // [16,6,5,5]
    const float* b2  = (const float*)d_in[4];
    const float* w3  = (const float*)d_in[5];   // [32,16,5,5]
    const float* b3  = (const float*)d_in[6];
    const float* wfc = (const float*)d_in[7];   // [10,512]
    const float* bfc = (const float*)d_in[8];
    float* outp = (float*)d_out;

    const int B = in_sizes[0] / (32 * 32);      // 8192

    // Workspace (floats): two ping-pong pooled buffers + weight panels.
    float* ws = (float*)d_ws;
    size_t off = 0;
    float* pA  = ws + off; off += (size_t)B * 6 * 16 * 16;   // pooled L1 / L3 out
    float* pB  = ws + off; off += (size_t)B * 16 * 8 * 8;    // pooled L2 out
    float* wq1 = ws + off; off += 28 * 16;                   // conv1 panel  (KPAD=28)
    float* wq2 = ws + off; off += 152 * 16;                  // conv2 panel  (KPAD=152)
    float* wq3 = ws + off; off += 2 * 400 * 16;              // conv3 panels (KPAD=400, 2 n-tiles)
    float* wqf = ws + off; off += 512 * 16;                  // fc panel

    // --- weight quantization + panel layout (tiny) ---
    prep_w_kernel<25, 6, 28>   <<<dim3((1 * 28 * 16  + 255) / 256), 256, 0, stream>>>(w1, wq1);
    prep_w_kernel<150, 16, 152><<<dim3((1 * 152 * 16 + 255) / 256), 256, 0, stream>>>(w2, wq2);
    prep_w_kernel<400, 32, 400><<<dim3((2 * 400 * 16 + 255) / 256), 256, 0, stream>>>(w3, wq3);
    prep_w_kernel<512, 10, 512><<<dim3((1 * 512 * 16 + 255) / 256), 256, 0, stream>>>(wfc, wqf);

    // --- layer 1: conv(1->6, 32x32) fused -> pooled 16x16 (quantizes x) ---
    conv_pool_wmma_kernel<1, 6, 32, 32, 28, true>
        <<<dim3(64, B, 1), 32, 0, stream>>>(x, wq1, b1, pA);

    // --- layer 2: conv(6->16, 16x16) fused -> pooled 8x8 ---
    conv_pool_wmma_kernel<6, 16, 16, 16, 152, false>
        <<<dim3(16, B, 1), 32, 0, stream>>>(pA, wq2, b2, pB);

    // --- layer 3: conv(16->32, 8x8) fused -> pooled 4x4 ---
    conv_pool_wmma_kernel<16, 32, 8, 8, 400, false>
        <<<dim3(4, B, 2), 32, 0, stream>>>(pB, wq3, b3, pA);

    // --- FC: [B,512] @ [512,10] + bias ---
    fc_wmma_kernel<<<dim3(B / 16), 32, 0, stream>>>(pA, wqf, bfc, outp);
}